// Model_7962869367673
// MI455X (gfx1250) — compile-verified
//
#include <hip/hip_runtime.h>

typedef __attribute__((ext_vector_type(2))) float v2f;
typedef __attribute__((ext_vector_type(8))) float v8f;

// ---------------------------------------------------------------- zero fill
__global__ void zero_f32(float* __restrict__ p, long n) {
  long i = (long)blockIdx.x * blockDim.x + threadIdx.x;
  long stride = (long)gridDim.x * blockDim.x;
  for (; i < n; i += stride) p[i] = 0.0f;
}

// ------------------------------------------------- edge scatter (mean prep)
// agg[dst] += feat[src] (elementwise, L2-resident float atomics), deg[dst] += 1
// One wave (32 lanes) per edge; each lane handles float4 chunks of the row.
__global__ void edge_scatter(const float* __restrict__ feat,
                             const int* __restrict__ src,
                             const int* __restrict__ dst,
                             float* __restrict__ agg,
                             float* __restrict__ deg,
                             int E, int d) {
  int warp   = (blockIdx.x * blockDim.x + threadIdx.x) >> 5;
  int lane   = threadIdx.x & 31;
  int nwarps = (gridDim.x * blockDim.x) >> 5;
  int chunks = d >> 2;
  for (int e = warp; e < E; e += nwarps) {
    int s = src[e], t = dst[e];
    const float4* xr = (const float4*)(feat + (long)s * d);
    float* ar = agg + (long)t * d;
    for (int c = lane; c < chunks; c += 32) {
      float4 v = xr[c];
      atomicAdd(ar + 4 * c + 0, v.x);
      atomicAdd(ar + 4 * c + 1, v.y);
      atomicAdd(ar + 4 * c + 2, v.z);
      atomicAdd(ar + 4 * c + 3, v.w);
    }
    if (lane == 0) atomicAdd(deg + t, 1.0f);
  }
}

// -------------------------------------------------- fused SAGE layer (WMMA)
// out[r, n] = act( sum_k x[r,k]*w_self[n,k] + mean[r,k]*w_neigh[n,k] + b[n] )
// Block = (Hpad/16) waves; one block per 16-row tile. x/mean tiles in LDS.
// Exact f32 path via V_WMMA_F32_16X16X4_F32.
template <int K>
__global__ __launch_bounds__(512)
void sage_layer(const float* __restrict__ x, const float* __restrict__ agg,
                const float* __restrict__ deg,
                const float* __restrict__ w_self,
                const float* __restrict__ w_neigh,
                const float* __restrict__ bias, float* __restrict__ out,
                int N, int Hpad, int Hvalid, int do_relu) {
  constexpr int LDK = K + 4;                 // LDS pad vs bank conflicts
  __shared__ float xs[16 * LDK];
  __shared__ float as[16 * LDK];

  const int r0 = blockIdx.x * 16;

  // Cooperative stage: 16 rows of x and agg/deg (mean) into LDS.
  for (int i = threadIdx.x; i < 16 * K; i += blockDim.x) {
    int r = i / K, c = i - r * K;            // K is power of two -> shifts
    int gr = r0 + r;
    float xv = 0.0f, av = 0.0f;
    if (gr < N) {
      xv = x[(long)gr * K + c];
      float dg = deg[gr];
      av = agg[(long)gr * K + c] / fmaxf(dg, 1.0f);
    }
    xs[r * LDK + c] = xv;
    as[r * LDK + c] = av;
  }
  __syncthreads();

  const int wave = threadIdx.x >> 5;         // wave32
  const int lane = threadIdx.x & 31;
  const int half = lane >> 4;                // K-half selector (ISA 7.12.2)
  const int lid  = lane & 15;                // output row (A) / col (B,C,D)
  const int n0   = wave * 16;
  if (n0 >= Hpad) return;                    // wave-uniform; EXEC stays full

  // Clamp weight-row for padded columns (never stored, avoids OOB reads).
  int nrow = n0 + lid;
  if (nrow >= Hvalid) nrow = Hvalid - 1;
  const float* wsr = w_self  + (long)nrow * K;
  const float* wnr = w_neigh + (long)nrow * K;

  v8f acc = {};
#pragma unroll
  for (int kk = 0; kk < K; kk += 4) {
    const int kb = kk + 2 * half;
    v2f aS, aN, bS, bN;
    aS.x = xs[lid * LDK + kb]; aS.y = xs[lid * LDK + kb + 1];
    aN.x = as[lid * LDK + kb]; aN.y = as[lid * LDK + kb + 1];
    bS.x = wsr[kb];            bS.y = wsr[kb + 1];
    bN.x = wnr[kb];            bN.y = wnr[kb + 1];
    // acc = aS*bS + (aN*bN + acc)
    acc = __builtin_amdgcn_wmma_f32_16x16x4_f32(false, aN, false, bN,
                                                (short)0, acc, false, false);
    acc = __builtin_amdgcn_wmma_f32_16x16x4_f32(false, aS, false, bS,
                                                (short)0, acc, false, false);
  }

  const int col = n0 + lid;
  const float bv = (col < Hvalid) ? bias[col] : 0.0f;
#pragma unroll
  for (int v = 0; v < 8; ++v) {
    int m = v + 8 * half;                    // D row per ISA layout
    int gr = r0 + m;
    float val = acc[v] + bv;
    if (do_relu) val = fmaxf(val, 0.0f);
    if (gr < N && col < Hvalid)
      out[(long)gr * Hvalid + col] = val;
  }
}

// -------------------------------------------------------------------- driver
extern "C" void kernel_launch(void* const* d_in, const int* in_sizes, int n_in,
                              void* d_out, int out_size, void* d_ws, size_t ws_size,
                              hipStream_t stream) {
  const float* x        = (const float*)d_in[0];
  const int*   src1     = (const int*)d_in[1];
  const int*   dst1     = (const int*)d_in[2];
  const int*   src2     = (const int*)d_in[3];
  const int*   dst2     = (const int*)d_in[4];
  const float* w_self1  = (const float*)d_in[5];
  const float* w_neigh1 = (const float*)d_in[6];
  const float* b1       = (const float*)d_in[7];
  const float* w_self2  = (const float*)d_in[8];
  const float* w_neigh2 = (const float*)d_in[9];
  const float* b2       = (const float*)d_in[10];
  float* out = (float*)d_out;

  const int IN = 128, H = 256;
  const int N  = in_sizes[0] / IN;
  const int E1 = in_sizes[1];
  const int E2 = in_sizes[3];
  const int C  = out_size / N;               // 40
  const int Cpad = ((C + 15) / 16) * 16;     // 48

  // Workspace layout (floats): deg1[N] deg2[N] h[N*H] aggbig[N*H]
  // agg1 (N*IN) aliases the front of aggbig; agg2 (N*H) reuses it after L1.
  float* ws   = (float*)d_ws;
  float* deg1 = ws;
  float* deg2 = ws + N;
  float* h    = ws + 2L * N;
  float* aggb = h + (long)N * H;
  float* agg1 = aggb;
  float* agg2 = aggb;

  const int ZT = 256;
  auto zblocks = [](long n) { return (int)((n + 255) / 256); };

  // ---- Layer 1 ----
  zero_f32<<<zblocks(N), ZT, 0, stream>>>(deg1, N);
  zero_f32<<<zblocks((long)N * IN), ZT, 0, stream>>>(agg1, (long)N * IN);

  {
    long threads = (long)E1 * 32;
    int blocks = (int)((threads + 255) / 256);
    edge_scatter<<<blocks, 256, 0, stream>>>(x, src1, dst1, agg1, deg1, E1, IN);
  }

  {
    int blocks = (N + 15) / 16;
    int tpb = (H / 16) * 32;                 // 512 threads = 16 waves
    sage_layer<IN><<<blocks, tpb, 0, stream>>>(x, agg1, deg1, w_self1, w_neigh1,
                                               b1, h, N, H, H, /*relu=*/1);
  }

  // ---- Layer 2 ----
  zero_f32<<<zblocks(N), ZT, 0, stream>>>(deg2, N);
  zero_f32<<<zblocks((long)N * H), ZT, 0, stream>>>(agg2, (long)N * H);

  {
    long threads = (long)E2 * 32;
    int blocks = (int)((threads + 255) / 256);
    edge_scatter<<<blocks, 256, 0, stream>>>(h, src2, dst2, agg2, deg2, E2, H);
  }

  {
    int blocks = (N + 15) / 16;
    int tpb = (Cpad / 16) * 32;              // 96 threads = 3 waves
    sage_layer<H><<<blocks, tpb, 0, stream>>>(h, agg2, deg2, w_self2, w_neigh2,
                                              b2, out, N, Cpad, C, /*relu=*/0);
  }
}